// GConv_20349555048807
// MI455X (gfx1250) — compile-verified
//
#include <hip/hip_runtime.h>
#include <stdint.h>

typedef float v2f __attribute__((ext_vector_type(2)));
typedef float v8f __attribute__((ext_vector_type(8)));

#define FD 128   // node feature dim D
#define FE 32    // edge feature dim DE
#define NET 5    // number of edge types

// ---------------------------------------------------------------------------
// CDNA5 async memory->LDS staging (ASYNCcnt-tracked, no VGPR round trip).
// lds_off: byte offset into workgroup LDS; gptr: per-lane global address.
// ---------------------------------------------------------------------------
__device__ __forceinline__ void async_load_b128_to_lds(unsigned lds_off,
                                                       const void* gptr) {
    unsigned long long ga = (unsigned long long)(uintptr_t)gptr;
    asm volatile("global_load_async_to_lds_b128 %0, %1, off"
                 :: "v"(lds_off), "v"(ga)
                 : "memory");
}
__device__ __forceinline__ void wait_async_lds() {
    asm volatile("s_wait_asynccnt 0" ::: "memory");
}

// ---------------------------------------------------------------------------
// Node-side GEMM: Xt = op_feats @ W[0:128,:] + bias
// Block = 256 threads = 8 waves; covers 64 rows x 128 cols.
// Wave w owns col tile [16w,16w+16) and keeps 4 accumulators (4 row tiles),
// so each B fragment load feeds 4 V_WMMA_F32_16X16X4_F32 ops.
// ---------------------------------------------------------------------------
__global__ void __launch_bounds__(256)
node_gemm_kernel(const float* __restrict__ X,     // [N,128]
                 const float* __restrict__ W,     // [160,128] (top 128 rows)
                 const float* __restrict__ bias,  // [128]
                 float* __restrict__ Xt,          // [N,128]
                 int nrows)
{
    __shared__ float As[64 * FD];                 // 32 KB A tile
    const int block_row = blockIdx.x * 64;
    const int tid = threadIdx.x;
    const unsigned lds_base = (unsigned)(uintptr_t)(&As[0]);

    // Stage 64x128 A tile: 2048 x 16B chunks, 8 async b128 loads per thread.
#pragma unroll
    for (int c = 0; c < 8; ++c) {
        int chunk = c * 256 + tid;                // 16B chunk id, 32 per row
        int r = block_row + (chunk >> 5);
        r = (r < nrows) ? r : (nrows - 1);        // clamp keeps lanes active
        const float* g = X + (size_t)r * FD + ((chunk & 31) << 2);
        async_load_b128_to_lds(lds_base + ((unsigned)chunk << 4), g);
    }
    wait_async_lds();
    __syncthreads();

    const int wave = tid >> 5;
    const int lane = tid & 31;
    const int colbase = wave << 4;
    const int rc   = lane & 15;                   // A row / B,C col
    const int hi   = lane >> 4;
    const int koff = hi << 1;                     // K offset 0 or 2

    v8f c[4] = {};
    for (int kb = 0; kb < FD; kb += 4) {
        v2f b;
        b.x = W[(size_t)(kb + koff)     * FD + colbase + rc];
        b.y = W[(size_t)(kb + koff + 1) * FD + colbase + rc];
#pragma unroll
        for (int t = 0; t < 4; ++t) {
            v2f a;
            a.x = As[(t * 16 + rc) * FD + kb + koff];
            a.y = As[(t * 16 + rc) * FD + kb + koff + 1];
            c[t] = __builtin_amdgcn_wmma_f32_16x16x4_f32(false, a, false, b,
                                                         (short)0, c[t],
                                                         false, false);
        }
    }

    const float bv = bias[colbase + rc];
#pragma unroll
    for (int t = 0; t < 4; ++t)
#pragma unroll
        for (int r = 0; r < 8; ++r) {
            int m = block_row + t * 16 + r + (hi << 3);   // C: M = r + 8*hi
            if (m < nrows)
                Xt[(size_t)m * FD + colbase + rc] = c[t][r] + bv;
        }
}

// ---------------------------------------------------------------------------
// Edge-side GEMM + scatter: m = ef @ W[128:160,:] + Xt[src]; S[dst] += m;
// cnt[dst] += 1.  Block covers 64 edges; wave w does col tile w with 4
// accumulators (4 edge tiles) per B fragment. K=32 -> 8 WMMA steps x 4 tiles.
// ---------------------------------------------------------------------------
__global__ void __launch_bounds__(256)
edge_gemm_scatter_kernel(const float* __restrict__ ef,   // [E,32]
                         const float* __restrict__ W,    // [160,128] (rows 128..159)
                         const float* __restrict__ Xt,   // [N,128] (bias folded)
                         const int* __restrict__ src,
                         const int* __restrict__ dst,
                         float* __restrict__ S,          // [N,128]
                         float* __restrict__ cnt,        // [N]
                         int E)
{
    __shared__ float Es[64 * FE];                 // 8 KB ef tile
    __shared__ int   ssrc[64];
    __shared__ int   sdst[64];

    const int e0  = blockIdx.x * 64;
    const int tid = threadIdx.x;
    const unsigned lds_base = (unsigned)(uintptr_t)(&Es[0]);

    // Stage 64x32 ef tile: 512 x 16B chunks, 2 async b128 loads per thread.
#pragma unroll
    for (int c = 0; c < 2; ++c) {
        int chunk = c * 256 + tid;                // 8 chunks per edge row
        int e = e0 + (chunk >> 3);
        e = (e < E) ? e : (E - 1);
        const float* g = ef + (size_t)e * FE + ((chunk & 7) << 2);
        async_load_b128_to_lds(lds_base + ((unsigned)chunk << 4), g);
    }
    if (tid < 64) {
        int e = e0 + tid;
        e = (e < E) ? e : (E - 1);
        ssrc[tid] = src[e];
        sdst[tid] = dst[e];
    }
    if (tid >= 64 && tid < 128) {                 // waves 2,3 bump counts
        int e = e0 + (tid - 64);
        if (e < E) unsafeAtomicAdd(&cnt[dst[e]], 1.0f);
    }
    wait_async_lds();
    __syncthreads();

    const int wave = tid >> 5;
    const int lane = tid & 31;
    const int colbase = wave << 4;
    const int rc   = lane & 15;
    const int hi   = lane >> 4;
    const int koff = hi << 1;

    v8f c[4] = {};
#pragma unroll
    for (int kb = 0; kb < FE; kb += 4) {
        v2f b;
        b.x = W[(size_t)(FD + kb + koff)     * FD + colbase + rc];
        b.y = W[(size_t)(FD + kb + koff + 1) * FD + colbase + rc];
#pragma unroll
        for (int t = 0; t < 4; ++t) {
            v2f a;
            a.x = Es[(t * 16 + rc) * FE + kb + koff];
            a.y = Es[(t * 16 + rc) * FE + kb + koff + 1];
            c[t] = __builtin_amdgcn_wmma_f32_16x16x4_f32(false, a, false, b,
                                                         (short)0, c[t],
                                                         false, false);
        }
    }

#pragma unroll
    for (int t = 0; t < 4; ++t)
#pragma unroll
        for (int r = 0; r < 8; ++r) {
            int m = t * 16 + r + (hi << 3);
            if (e0 + m < E) {
                float v = c[t][r] + Xt[(size_t)ssrc[m] * FD + colbase + rc];
                unsafeAtomicAdd(&S[(size_t)sdst[m] * FD + colbase + rc], v);
            }
        }
}

// ---------------------------------------------------------------------------
// acc += (cnt > 0) ? S / cnt : 0      (per-etype mean)
// ---------------------------------------------------------------------------
__global__ void __launch_bounds__(256)
norm_acc_kernel(const float* __restrict__ S, const float* __restrict__ cnt,
                float* __restrict__ acc, int total)
{
    int i = blockIdx.x * 256 + threadIdx.x;
    if (i < total) {
        float cv = cnt[i >> 7];
        if (cv > 0.0f) acc[i] += S[i] / cv;
    }
}

// ---------------------------------------------------------------------------
// out = relu(op_feats + acc / 5)
// ---------------------------------------------------------------------------
__global__ void __launch_bounds__(256)
finalize_kernel(const float* __restrict__ X, float* __restrict__ out, int total)
{
    int i = blockIdx.x * 256 + threadIdx.x;
    if (i < total) {
        float v = X[i] + out[i] * 0.2f;
        out[i] = v > 0.0f ? v : 0.0f;
    }
}

extern "C" void kernel_launch(void* const* d_in, const int* in_sizes, int n_in,
                              void* d_out, int out_size, void* d_ws, size_t ws_size,
                              hipStream_t stream) {
    (void)n_in; (void)out_size; (void)ws_size;

    const float* op = (const float*)d_in[0];
    const int N = in_sizes[0] / FD;

    float* out = (float*)d_out;
    // workspace: S [N*FD] | cnt [N] | Xt [N*FD]
    float* S   = (float*)d_ws;
    float* cnt = S + (size_t)N * FD;
    float* Xt  = cnt + N;

    hipMemsetAsync(out, 0, (size_t)N * FD * sizeof(float), stream);

    for (int et = 0; et < NET; ++et) {
        const float* ef   = (const float*)d_in[1 + 5 * et];
        const float* W    = (const float*)d_in[2 + 5 * et];
        const float* bias = (const float*)d_in[3 + 5 * et];
        const int*   src  = (const int*)  d_in[4 + 5 * et];
        const int*   dst  = (const int*)  d_in[5 + 5 * et];
        const int E = in_sizes[4 + 5 * et];

        hipMemsetAsync(S, 0, ((size_t)N * FD + N) * sizeof(float), stream);

        node_gemm_kernel<<<(N + 63) / 64, 256, 0, stream>>>(op, W, bias, Xt, N);
        edge_gemm_scatter_kernel<<<(E + 63) / 64, 256, 0, stream>>>(
            ef, W, Xt, src, dst, S, cnt, E);
        norm_acc_kernel<<<((size_t)N * FD + 255) / 256, 256, 0, stream>>>(
            S, cnt, out, N * FD);
    }

    finalize_kernel<<<((size_t)N * FD + 255) / 256, 256, 0, stream>>>(op, out, N * FD);
}